// HungarianMatcher_88957362635394
// MI455X (gfx1250) — compile-verified
//
#include <hip/hip_runtime.h>

// Problem constants (match reference)
#define B_   2
#define Q_   300
#define M_   20
#define C_   80
#define P_   32
#define T_   2
#define THW_ 32768
#define M2_  32          // padded M for dot workspaces (two 16-wide WMMA N-tiles)
#define CHUNKS_ 16
#define TILES_  (THW_ / 16 / CHUNKS_)   // 128 j-tiles per chunk
#define NCAND_ 10

#define LOG2E_ 1.4426950408889634f
#define LN2_   0.6931471805599453f

typedef __attribute__((ext_vector_type(2))) float v2f;
typedef __attribute__((ext_vector_type(8))) float v8f;

static __device__ __forceinline__ v8f wmma4(v2f a, v2f b, v8f c) {
  // D = A(16x4 f32) x B(4x16 f32) + C(16x16 f32)
  return __builtin_amdgcn_wmma_f32_16x16x4_f32(false, a, false, b, (short)0, c,
                                               false, false);
}

// ---------------------------------------------------------------------------
// Kernel 0: zero the atomic-accumulated workspace regions
// ---------------------------------------------------------------------------
__global__ void zero_kernel(float* __restrict__ p, int n) {
  int i = blockIdx.x * blockDim.x + threadIdx.x;
  if (i < n) p[i] = 0.0f;
}

// ---------------------------------------------------------------------------
// Kernel 1: fused mask-cost GEMMs via WMMA f32 16x16x4.
//   X = coeff @ proto (per 16q x 16j tile), then softplus/sigmoid elementwise,
//   then accumulate X@t^T and sigmoid(X)@t^T through an LDS re-layout.
// grid = (CHUNKS_, ceil(Q/16), B), block = 32 (one wave)
// ---------------------------------------------------------------------------
__global__ __launch_bounds__(32) void mask_cost_kernel(
    const float* __restrict__ coeffG, const float* __restrict__ protoG,
    const float* __restrict__ tmaskG,
    float* __restrict__ dotX, float* __restrict__ dotS,
    float* __restrict__ sspG, float* __restrict__ ssgG)
{
  const int lane = threadIdx.x;
  const int half = lane >> 4;     // 0: lanes 0-15, 1: lanes 16-31
  const int l16  = lane & 15;
  const int b    = blockIdx.z;
  const int q0   = blockIdx.y * 16;
  const int chunk = blockIdx.x;

  const float* cf = coeffG + b * Q_ * P_;
  const float* pr = protoG + b * P_ * THW_;
  const float* tm = tmaskG + b * M_ * THW_;

  // Preload coeff block [16q x 32p] in WMMA A-layout:
  // lane row M = l16; k-step ks covers p = 4*ks..4*ks+3, lane half picks {p,p+1} vs {p+2,p+3}.
  v2f a[8];
  {
    int qr = q0 + l16; if (qr >= Q_) qr = Q_ - 1;   // clamp; out rows dropped at write
    #pragma unroll
    for (int ks = 0; ks < 8; ++ks) {
      int p = ks * 4 + 2 * half;
      a[ks] = *reinterpret_cast<const v2f*>(cf + qr * P_ + p);
    }
  }

  v8f cx0 = {}, cx1 = {}, cs0 = {}, cs1 = {};
  float ssp[8], ssg[8];
  #pragma unroll
  for (int r = 0; r < 8; ++r) { ssp[r] = 0.0f; ssg[r] = 0.0f; }

  __shared__ float ldsX[256];   // 16q x 16j tile of x
  __shared__ float ldsS[256];   // 16q x 16j tile of sigmoid(x)

  for (int tile = 0; tile < TILES_; ++tile) {
    const int j0 = (chunk * TILES_ + tile) * 16;
    const int j  = j0 + l16;

    if (tile + 1 < TILES_) __builtin_prefetch(tm + l16 * THW_ + j0 + 16, 0, 1);

    // ---- GEMM1: X = coeff @ proto, K = P = 32 (8 wmma k-steps)
    v8f x = {};
    #pragma unroll
    for (int ks = 0; ks < 8; ++ks) {
      int p = ks * 4 + 2 * half;       // B-layout: VGPR0 row k (half0) / k+2 (half1)
      v2f bv;
      bv.x = pr[p * THW_ + j];
      bv.y = pr[(p + 1) * THW_ + j];
      x = wmma4(a[ks], bv, x);
    }

    // ---- elementwise + per-q partial sums; stash tiles in LDS (q-major)
    // Raw hardware transcendentals (args are range-safe, no fixup needed):
    //   e = exp(-|x|) = exp2(-|x|*log2e)   in (0,1], underflow->0 is exact enough
    //   d = 1+e in (1,2]: v_log_f32 (log2) needs no denorm/inf guards
    //   softplus(x) = max(x,0) + ln2*log2(d)
    //   sigmoid(x)  = x>=0 ? 1/d : 1 - 1/d     (v_rcp_f32)
    #pragma unroll
    for (int r = 0; r < 8; ++r) {
      float xv = x[r];                                   // (q = r+8*half, j = l16)
      float e  = __builtin_amdgcn_exp2f(-fabsf(xv) * LOG2E_);
      float d  = 1.0f + e;
      float rc = __builtin_amdgcn_rcpf(d);               // v_rcp_f32
      float sp = fmaf(LN2_, __builtin_amdgcn_logf(d), fmaxf(xv, 0.0f));
      float sg = (xv >= 0.0f) ? rc : (1.0f - rc);
      ssp[r] += sp;
      ssg[r] += sg;
      ldsX[(r + 8 * half) * 16 + l16] = xv;
      ldsS[(r + 8 * half) * 16 + l16] = sg;
    }
    // LDS ops are in-order within a wave; compiler inserts s_wait_dscnt.

    // ---- GEMM2: C[q,m] += X_tile @ t^T (and sigmoid tile), K = 16 j's (4 k-steps)
    #pragma unroll
    for (int ks = 0; ks < 4; ++ks) {
      int jj = ks * 4 + 2 * half;      // lane-half K offset for A and B layouts
      v2f ax = *reinterpret_cast<const v2f*>(&ldsX[l16 * 16 + jj]);
      v2f as = *reinterpret_cast<const v2f*>(&ldsS[l16 * 16 + jj]);
      v2f b0 = *reinterpret_cast<const v2f*>(tm + l16 * THW_ + j0 + jj);
      v2f b1 = {0.0f, 0.0f};
      if (16 + l16 < M_)
        b1 = *reinterpret_cast<const v2f*>(tm + (16 + l16) * THW_ + j0 + jj);
      cx0 = wmma4(ax, b0, cx0);
      cx1 = wmma4(ax, b1, cx1);
      cs0 = wmma4(as, b0, cs0);
      cs1 = wmma4(as, b1, cs1);
    }
  }

  // ---- per-q softplus/sigmoid sums: reduce across 16 lanes of each half
  #pragma unroll
  for (int r = 0; r < 8; ++r) {
    float vsp = ssp[r], vsg = ssg[r];
    #pragma unroll
    for (int mk = 8; mk >= 1; mk >>= 1) {
      vsp += __shfl_xor(vsp, mk, 32);
      vsg += __shfl_xor(vsg, mk, 32);
    }
    int q = q0 + r + 8 * half;
    if (l16 == 0 && q < Q_) {
      atomicAdd(&sspG[b * Q_ + q], vsp);
      atomicAdd(&ssgG[b * Q_ + q], vsg);
    }
  }
  // ---- dot accumulators: D layout (row q = r+8*half, col m = l16)
  #pragma unroll
  for (int r = 0; r < 8; ++r) {
    int q = q0 + r + 8 * half;
    if (q < Q_) {
      atomicAdd(&dotX[(b * Q_ + q) * M2_ + l16], cx0[r]);
      atomicAdd(&dotS[(b * Q_ + q) * M2_ + l16], cs0[r]);
      if (16 + l16 < M_) {
        atomicAdd(&dotX[(b * Q_ + q) * M2_ + 16 + l16], cx1[r]);
        atomicAdd(&dotS[(b * Q_ + q) * M2_ + 16 + l16], cs1[r]);
      }
    }
  }
}

// ---------------------------------------------------------------------------
// Kernel 2: t_sum[b][m] = sum over THW of match_masks
// ---------------------------------------------------------------------------
__global__ __launch_bounds__(256) void tsum_kernel(const float* __restrict__ tm,
                                                   float* __restrict__ out) {
  __shared__ float red[256];
  const int m = blockIdx.x, b = blockIdx.y, tid = threadIdx.x;
  const float* p = tm + (b * M_ + m) * THW_;
  float s = 0.0f;
  for (int i = tid; i < THW_; i += 256) s += p[i];
  red[tid] = s; __syncthreads();
  for (int st = 128; st > 0; st >>= 1) {
    if (tid < st) red[tid] += red[tid + st];
    __syncthreads();
  }
  if (tid == 0) out[b * M_ + m] = red[0];
}

// ---------------------------------------------------------------------------
// Kernel 3: assemble Cost[b,q,m] and giou[b,q,m]
// ---------------------------------------------------------------------------
__global__ __launch_bounds__(64) void cost_kernel(
    const float* __restrict__ cls, const float* __restrict__ boxes,
    const int* __restrict__ labels, const float* __restrict__ gtb,
    const float* __restrict__ dotX, const float* __restrict__ dotS,
    const float* __restrict__ ssp, const float* __restrict__ ssg,
    const float* __restrict__ tsum,
    float* __restrict__ giouOut, float* __restrict__ out)
{
  int q = blockIdx.x * blockDim.x + threadIdx.x;
  int b = blockIdx.y;
  if (q >= Q_) return;

  float ob[T_][4];
  #pragma unroll
  for (int t = 0; t < T_; ++t)
    #pragma unroll
    for (int k = 0; k < 4; ++k)
      ob[t][k] = boxes[((b * Q_ + q) * T_ + t) * 4 + k];

  // anchor-in-box penalty: per frame, any gt (box or center region)
  const float rr = 2.5f / 32.0f;
  int outCnt = 0;
  #pragma unroll
  for (int t = 0; t < T_; ++t) {
    float cx = (ob[t][0] + ob[t][2]) * 0.5f;
    float cy = (ob[t][1] + ob[t][3]) * 0.5f;
    bool inAny = false;
    for (int m = 0; m < M_; ++m) {
      const float* g = gtb + ((b * M_ + m) * T_ + t) * 4;
      bool in_b = (cx > g[0]) && (cx < g[2]) && (cy > g[1]) && (cy < g[3]);
      float gcx = (g[0] + g[2]) * 0.5f;
      float gcy = (g[1] + g[3]) * 0.5f;
      bool in_c = (cx > gcx - rr) && (cx < gcx + rr) &&
                  (cy > gcy - rr) && (cy < gcy + rr);
      inAny = inAny || in_b || in_c;
    }
    if (!inAny) outCnt++;
  }

  const float sspq = ssp[b * Q_ + q];
  const float ssgq = ssg[b * Q_ + q];

  for (int m = 0; m < M_; ++m) {
    int lbl = labels[b * M_ + m];
    float logit = cls[(b * Q_ + q) * C_ + lbl];
    float cost_class = -(1.0f / (1.0f + __expf(-logit)));

    const float* g = gtb + (b * M_ + m) * T_ * 4;
    float l1 = 0.0f, gsum = 0.0f, vcnt = 0.0f;
    #pragma unroll
    for (int t = 0; t < T_; ++t) {
      const float* gt = g + t * 4;
      #pragma unroll
      for (int k = 0; k < 4; ++k) l1 += fabsf(ob[t][k] - gt[k]);
      float a1 = (ob[t][2] - ob[t][0]) * (ob[t][3] - ob[t][1]);
      float a2 = (gt[2] - gt[0]) * (gt[3] - gt[1]);
      float iw = fmaxf(fminf(ob[t][2], gt[2]) - fmaxf(ob[t][0], gt[0]), 0.0f);
      float ih = fmaxf(fminf(ob[t][3], gt[3]) - fmaxf(ob[t][1], gt[1]), 0.0f);
      float inter = iw * ih;
      float uni = a1 + a2 - inter;
      float iou = inter / (uni + 1e-7f);
      float ew = fmaxf(fmaxf(ob[t][2], gt[2]) - fminf(ob[t][0], gt[0]), 0.0f);
      float eh = fmaxf(fmaxf(ob[t][3], gt[3]) - fminf(ob[t][1], gt[1]), 0.0f);
      float enc = ew * eh;
      float gv = iou - (enc - uni) / (enc + 1e-7f);
      bool valid = ((gt[2] - gt[0]) > 0.0f) && ((gt[3] - gt[1]) > 0.0f);
      if (valid) { gsum += gv; vcnt += 1.0f; }
    }
    float giou = gsum / fmaxf(vcnt, 1.0f);
    giouOut[(b * Q_ + q) * M_ + m] = giou;

    // cost_bce*THW = Ssp - <t, x>   (softplus(-x) = softplus(x) - x)
    float bce = (sspq - dotX[(b * Q_ + q) * M2_ + m]) * (1.0f / (float)THW_);
    float dn  = dotS[(b * Q_ + q) * M2_ + m];
    float dice = 1.0f - (2.0f * dn + 1.0f) / (ssgq + tsum[b * M_ + m] + 1.0f);

    float Cm = cost_class + 3.0f * (l1 + 1.0f - giou) + (bce + dice);
    if (isnan(Cm) || isinf(Cm)) Cm = 1000.0f;
    out[(b * Q_ + q) * M_ + m] = Cm + 100.0f * (float)outCnt;
  }
}

// ---------------------------------------------------------------------------
// Kernel 4: dynamic-k matching, one block per sample
// ---------------------------------------------------------------------------
static __device__ void resolve_conflicts(float* cst, float* mm, int tid) {
  if (tid < Q_) {
    float rs = 0.0f;
    for (int m = 0; m < M_; ++m) rs += mm[tid * M_ + m];
    if (rs > 1.0f) {
      float bv = 3.4e38f; int bi = 0;
      for (int m = 0; m < M_; ++m) {
        float v = cst[tid * M_ + m];
        if (v < bv) { bv = v; bi = m; }
      }
      for (int m = 0; m < M_; ++m) mm[tid * M_ + m] = (m == bi) ? 1.0f : 0.0f;
    }
  }
}

__global__ __launch_bounds__(320) void match_kernel(
    const float* __restrict__ giouG, float* __restrict__ d_out)
{
  const int b = blockIdx.x, tid = threadIdx.x;
  __shared__ float cst[Q_ * M_];
  __shared__ float mm[Q_ * M_];
  __shared__ int flags[2];

  const float* costG = d_out + b * Q_ * M_;
  float* mmG = d_out + B_ * Q_ * M_ + b * Q_ * M_;
  const float* iouG = giouG + b * Q_ * M_;

  for (int i = tid; i < Q_ * M_; i += blockDim.x) {
    cst[i] = costG[i];
    mm[i] = 0.0f;
  }
  __syncthreads();

  if (tid < M_) {
    // dynamic_k: sum of top-10 clipped ious of this column
    int used[NCAND_];
    float s = 0.0f;
    for (int k = 0; k < NCAND_; ++k) {
      float bv = -1.0f; int bi = 0;
      for (int q = 0; q < Q_; ++q) {
        bool skip = false;
        for (int j = 0; j < k; ++j) if (used[j] == q) { skip = true; break; }
        if (skip) continue;
        float v = fmaxf(iouG[q * M_ + tid], 0.0f);
        if (v > bv) { bv = v; bi = q; }
      }
      used[k] = bi; s += bv;
    }
    int dk = (int)s; if (dk < 2) dk = 2;
    // 10 smallest costs of this column (ascending), set first dk
    int idx[NCAND_];
    for (int k = 0; k < NCAND_; ++k) {
      float bv = 3.4e38f; int bi = 0;
      for (int q = 0; q < Q_; ++q) {
        bool skip = false;
        for (int j = 0; j < k; ++j) if (idx[j] == q) { skip = true; break; }
        if (skip) continue;
        float v = cst[q * M_ + tid];
        if (v < bv) { bv = v; bi = q; }
      }
      idx[k] = bi;
    }
    for (int k = 0; k < NCAND_; ++k)
      if (k < dk) mm[idx[k] * M_ + tid] = 1.0f;
  }
  __syncthreads();
  resolve_conflicts(cst, mm, tid);
  __syncthreads();

  for (int it = 0; it < M_; ++it) {
    if (tid == 0) { flags[0] = 0; flags[1] = 0; }
    __syncthreads();
    if (tid < Q_) {
      float rs = 0.0f;
      for (int m = 0; m < M_; ++m) rs += mm[tid * M_ + m];
      if (rs == 0.0f) flags[1] = 1;          // some row unmatched
    }
    if (tid < M_) {
      float csum = 0.0f;
      for (int q = 0; q < Q_; ++q) csum += mm[q * M_ + tid];
      if (csum == 0.0f) flags[0] = 1;        // some gt unmatched
    }
    __syncthreads();
    if (!(flags[0] && flags[1])) break;

    if (tid < Q_) {
      float rs = 0.0f;
      for (int m = 0; m < M_; ++m) rs += mm[tid * M_ + m];
      if (rs > 0.0f)
        for (int m = 0; m < M_; ++m) cst[tid * M_ + m] += 100000.0f;
    }
    __syncthreads();
    if (tid < M_) {
      float csum = 0.0f;
      for (int q = 0; q < Q_; ++q) csum += mm[q * M_ + tid];
      if (csum == 0.0f) {
        float bv = 3.4e38f; int bi = 0;
        for (int q = 0; q < Q_; ++q) {
          float v = cst[q * M_ + tid];
          if (v < bv) { bv = v; bi = q; }
        }
        mm[bi * M_ + tid] = 1.0f;
      }
    }
    __syncthreads();
    resolve_conflicts(cst, mm, tid);
    __syncthreads();
  }

  for (int i = tid; i < Q_ * M_; i += blockDim.x) mmG[i] = mm[i];
}

// ---------------------------------------------------------------------------
extern "C" void kernel_launch(void* const* d_in, const int* in_sizes, int n_in,
                              void* d_out, int out_size, void* d_ws, size_t ws_size,
                              hipStream_t stream) {
  const float* cls    = (const float*)d_in[0];
  const float* coeff  = (const float*)d_in[1];
  const float* proto  = (const float*)d_in[2];
  const float* boxes  = (const float*)d_in[3];
  const int*   labels = (const int*)d_in[4];
  const float* masks  = (const float*)d_in[5];
  const float* gtb    = (const float*)d_in[6];
  float* out = (float*)d_out;

  // workspace layout (floats): ~206 KB total
  float* ws      = (float*)d_ws;
  float* wsDotX  = ws;                              // B*Q*32
  float* wsDotS  = wsDotX + B_ * Q_ * M2_;          // B*Q*32
  float* wsSsp   = wsDotS + B_ * Q_ * M2_;          // B*Q
  float* wsSsg   = wsSsp + B_ * Q_;                 // B*Q
  float* wsTsum  = wsSsg + B_ * Q_;                 // B*M
  float* wsGiou  = wsTsum + B_ * M_;                // B*Q*M

  const int nz = 2 * B_ * Q_ * M2_ + 2 * B_ * Q_;   // atomic-accumulated regions
  zero_kernel<<<(nz + 255) / 256, 256, 0, stream>>>(ws, nz);

  mask_cost_kernel<<<dim3(CHUNKS_, (Q_ + 15) / 16, B_), 32, 0, stream>>>(
      coeff, proto, masks, wsDotX, wsDotS, wsSsp, wsSsg);

  tsum_kernel<<<dim3(M_, B_), 256, 0, stream>>>(masks, wsTsum);

  cost_kernel<<<dim3((Q_ + 63) / 64, B_), 64, 0, stream>>>(
      cls, boxes, labels, gtb, wsDotX, wsDotS, wsSsp, wsSsg, wsTsum, wsGiou, out);

  match_kernel<<<B_, 320, 0, stream>>>(wsGiou, out);
}